// QCNN_58025008169536
// MI455X (gfx1250) — compile-verified
//
#include <hip/hip_runtime.h>
#include <math.h>

typedef float v2f __attribute__((ext_vector_type(2)));
typedef float v8f __attribute__((ext_vector_type(8)));

#define QPI 3.14159265358979323846f

struct Cx { float x, y; };
__device__ __forceinline__ Cx cxmul(Cx a, Cx b) { return Cx{a.x*b.x - a.y*b.y, a.x*b.y + a.y*b.x}; }
__device__ __forceinline__ Cx cxadd(Cx a, Cx b) { return Cx{a.x + b.x, a.y + b.y}; }

// ---------------------------------------------------------------------------
// Prep kernel (1 block, 32 threads): the parameterized 2-layer block is a
// FIXED 16x16 complex unitary W (weights shared across all states). Thread j
// simulates basis column |j> through both layers (RX RY RZ per qubit + CZ
// masks) and writes the real 32x32 matrix [[Wr,-Wi],[Wi,Wr]] row-major into
// workspace.
// ---------------------------------------------------------------------------
__global__ void qcnn_prep_kernel(const float* __restrict__ w, float* __restrict__ M) {
  int j = threadIdx.x;
  if (j >= 16) return;
  Cx a[16];
  for (int b = 0; b < 16; ++b) a[b] = Cx{ (b == j) ? 1.f : 0.f, 0.f };
  for (int h = 0; h < 2; ++h) {
    for (int q = 0; q < 4; ++q) {
      const float* wp = w + (h * 4 + q) * 3;
      float ca = cosf(0.5f * wp[0]), sa = sinf(0.5f * wp[0]);
      float cb = cosf(0.5f * wp[1]), sb = sinf(0.5f * wp[1]);
      float cc = cosf(0.5f * wp[2]), sc = sinf(0.5f * wp[2]);
      // Rx(a)=[[ca,-i sa],[-i sa,ca]]  Ry(b)=[[cb,-sb],[sb,cb]]  Rz(c)=diag(e^{-ic/2},e^{ic/2})
      Cx Rx[2][2] = {{{ca,0.f},{0.f,-sa}}, {{0.f,-sa},{ca,0.f}}};
      Cx Ry[2][2] = {{{cb,0.f},{-sb,0.f}}, {{sb,0.f},{cb,0.f}}};
      Cx Rz[2][2] = {{{cc,-sc},{0.f,0.f}}, {{0.f,0.f},{cc,sc}}};
      Cx T[2][2], G[2][2];
      for (int r = 0; r < 2; ++r)
        for (int c = 0; c < 2; ++c)
          T[r][c] = cxadd(cxmul(Ry[r][0], Rx[0][c]), cxmul(Ry[r][1], Rx[1][c]));
      for (int r = 0; r < 2; ++r)
        for (int c = 0; c < 2; ++c)
          G[r][c] = cxadd(cxmul(Rz[r][0], T[0][c]), cxmul(Rz[r][1], T[1][c]));
      int mask = 8 >> q;  // qubit q occupies bit (3-q) of the amplitude index
      for (int b = 0; b < 16; ++b) {
        if (b & mask) continue;
        int b1 = b | mask;
        Cx a0 = a[b], a1 = a[b1];
        a[b]  = cxadd(cxmul(G[0][0], a0), cxmul(G[0][1], a1));
        a[b1] = cxadd(cxmul(G[1][0], a0), cxmul(G[1][1], a1));
      }
    }
    for (int b = 0; b < 16; ++b) {          // CZ(0,1), CZ(2,3), CZ(1,2)
      float s = 1.f;
      if ((b & 0xC) == 0xC) s = -s;
      if ((b & 0x3) == 0x3) s = -s;
      if ((b & 0x6) == 0x6) s = -s;
      a[b].x *= s; a[b].y *= s;
    }
  }
  for (int r = 0; r < 16; ++r) {
    M[r * 32 + j]             =  a[r].x;   // Wr
    M[r * 32 + 16 + j]        = -a[r].y;   // -Wi
    M[(16 + r) * 32 + j]      =  a[r].y;   // Wi
    M[(16 + r) * 32 + 16 + j] =  a[r].x;   // Wr
  }
}

// ---------------------------------------------------------------------------
// Main kernel: one wave handles 16 states (col = lane&15). The B-fragment
// layout needs lane l to supply X rows with (row mod 4) in {2*half, 2*half+1},
// which is exactly the amplitudes whose qubit-2 bit equals `half`. So the two
// half-waves compute DISJOINT halves of each state: lane l builds only the 8
// amplitudes v2[half] (x) v0 (x) v1 (x) v3 of state (l&15) — 14 complex muls,
// constant indexing, zero redundancy, no cndmask selects. Then
// D(32x16) = M(32x32) * X(32x16) via 16 chained v_wmma_f32_16x16x4_f32,
// |amp|^2 and per-qubit Z reductions (one cross-half shfl_xor).
// ---------------------------------------------------------------------------
__global__ __launch_bounds__(256) void qcnn_main_kernel(const float* __restrict__ x,
                                                        const float* __restrict__ M,
                                                        float* __restrict__ out,
                                                        int ntiles) {
  const int lane = threadIdx.x & 31;
  const int wid  = (blockIdx.x * (blockDim.x >> 5)) + (threadIdx.x >> 5);
  if (wid >= ntiles) return;                 // uniform per wave: EXEC stays all-1s
  const int half = lane >> 4;
  const int col  = lane & 15;
  const int state = wid * 16 + col;

  // A-fragments (ISA 7.12.2, f32 A 16x4): vgpr0 = K(2*half), vgpr1 = K(2*half+1)
  v2f afrag0[8], afrag1[8];
  {
    const float* rowp0 = M + (size_t)col * 32 + 2 * half;          // Yre rows 0..15
    const float* rowp1 = M + (size_t)(16 + col) * 32 + 2 * half;   // Yim rows 16..31
#pragma unroll
    for (int f = 0; f < 8; ++f) {
      afrag0[f] = *(const v2f*)(rowp0 + 4 * f);
      afrag1[f] = *(const v2f*)(rowp1 + 4 * f);
    }
  }

  // Angle encoding: RZ(pi x0) is a global phase -> dropped.
  // Per qubit: v[0] = (cy*cz, -cy*sz), v[1] = (sy*cz, sy*sz),
  //            cy = cos(pi x1 / 2) etc.
  const float* xs = x + (size_t)state * 12;
  const float4 f0 = *(const float4*)(xs + 0);
  const float4 f1 = *(const float4*)(xs + 4);
  const float4 f2 = *(const float4*)(xs + 8);
  const float xy[4] = {f0.y, f1.x, f1.w, f2.z};   // x[...,1]
  const float xz[4] = {f0.z, f1.y, f2.x, f2.w};   // x[...,2]

  float cy[4], sy[4], cz[4], sz[4];
#pragma unroll
  for (int q = 0; q < 4; ++q) {
    float ay = 0.5f * QPI * xy[q];
    float az = 0.5f * QPI * xz[q];
    cy[q] = __cosf(ay); sy[q] = __sinf(ay);
    cz[q] = __cosf(az); sz[q] = __sinf(az);
  }

  // Qubit-2 factor for this half-wave, folded in as the tensor-product seed.
  float t2 = half ? sy[2] : cy[2];
  float u2 = t2 * sz[2];
  Cx seed = Cx{ t2 * cz[2], half ? u2 : -u2 };

  // a8[m], m = q0*4 + q1*2 + q3 : expand qubits 0, 1, 3 around the seed.
  Cx a8[8];
  a8[0] = seed;
#pragma unroll
  for (int e = 0; e < 3; ++e) {
    const int q = (e == 2) ? 3 : e;          // qubit order 0, 1, 3
    Cx v0 = Cx{ cy[q] * cz[q], -cy[q] * sz[q] };
    Cx v1 = Cx{ sy[q] * cz[q],  sy[q] * sz[q] };
#pragma unroll
    for (int i = (1 << e) - 1; i >= 0; --i) {
      Cx t = a8[i];
      a8[2 * i + 1] = cxmul(t, v1);
      a8[2 * i]     = cxmul(t, v0);
    }
  }

  // B-fragments: frag f covers X rows 4f + 2*half + {0,1}.
  //   f < 4  -> Re(amp b), b = 4f + 2*half + j  -> a8[2f + j].x
  //   f >= 4 -> Im(amp b), b = 4(f-4) + 2*half + j -> a8[2(f-4) + j].y
  v2f bfrag[8];
#pragma unroll
  for (int f = 0; f < 4; ++f) {
    bfrag[f].x     = a8[2 * f].x;
    bfrag[f].y     = a8[2 * f + 1].x;
    bfrag[f + 4].x = a8[2 * f].y;
    bfrag[f + 4].y = a8[2 * f + 1].y;
  }

  v8f acc0 = {};   // Yre (rows 0..15)
  v8f acc1 = {};   // Yim (rows 16..31)
#pragma unroll
  for (int f = 0; f < 8; ++f) {
    acc0 = __builtin_amdgcn_wmma_f32_16x16x4_f32(false, afrag0[f], false, bfrag[f],
                                                 (short)0, acc0, false, false);
    acc1 = __builtin_amdgcn_wmma_f32_16x16x4_f32(false, afrag1[f], false, bfrag[f],
                                                 (short)0, acc1, false, false);
  }

  // C/D layout: lane holds amplitude indices p = v + 8*half, state = lane&15.
  // Z_q sign: q0 <- bit3 (half), q1 <- bit2 (v>>2), q2 <- bit1, q3 <- bit0.
  float sA = 0.f, s1 = 0.f, s2 = 0.f, s3 = 0.f;
#pragma unroll
  for (int v = 0; v < 8; ++v) {
    float p = acc0[v] * acc0[v] + acc1[v] * acc1[v];
    sA += p;
    s1 = ((v >> 2) & 1) ? s1 - p : s1 + p;
    s2 = ((v >> 1) & 1) ? s2 - p : s2 + p;
    s3 = (v & 1)        ? s3 - p : s3 + p;
  }
  float oA = __shfl_xor(sA, 16, 32);
  float o1 = __shfl_xor(s1, 16, 32);
  float o2 = __shfl_xor(s2, 16, 32);
  float o3 = __shfl_xor(s3, 16, 32);
  if (half == 0) {
    float4 z;
    z.x = sA - oA;   // low half holds p<8 (bit3 = 0)
    z.y = s1 + o1;
    z.z = s2 + o2;
    z.w = s3 + o3;
    *(float4*)(out + (size_t)state * 4) = z;
  }
}

// ---------------------------------------------------------------------------
extern "C" void kernel_launch(void* const* d_in, const int* in_sizes, int n_in,
                              void* d_out, int out_size, void* d_ws, size_t ws_size,
                              hipStream_t stream) {
  (void)n_in; (void)out_size; (void)ws_size;
  const float* x = (const float*)d_in[0];   // (128,2048,4,3) f32
  const float* w = (const float*)d_in[1];   // (2,4,3) f32
  float* out = (float*)d_out;               // (128,2048,4) f32
  float* M   = (float*)d_ws;                // 32*32 f32 = 4 KB

  const int nstate = in_sizes[0] / 12;      // 262144
  const int ntiles = (nstate + 15) / 16;    // states per wave = 16

  qcnn_prep_kernel<<<1, 32, 0, stream>>>(w, M);

  const int wavesPerBlock = 8;              // 256 threads
  const int blocks = (ntiles + wavesPerBlock - 1) / wavesPerBlock;
  qcnn_main_kernel<<<blocks, wavesPerBlock * 32, 0, stream>>>(x, M, out, ntiles);
}